// AnomalyAwareModel_7103875908244
// MI455X (gfx1250) — compile-verified
//
#include <hip/hip_runtime.h>
#include <hip/hip_bf16.h>

typedef __attribute__((ext_vector_type(2))) float v2f;
typedef __attribute__((ext_vector_type(8))) float v8f;

static constexpr int SEM = 384, HID = 64, ALIGN = 32;

// ---------------------------------------------------------------- utilities
__device__ inline void atomicMaxF(float* addr, float val) {
    int old = __float_as_int(*addr);
    while (__int_as_float(old) < val) {
        int assumed = old;
        old = atomicCAS((int*)addr, assumed, __float_as_int(val));
        if (old == assumed) break;
    }
}

// ---------------------------------------------------------------- init
__global__ void k_init(float* deg, float* zg, float* segmax, float* denom,
                       int N, int B) {
    int i = blockIdx.x * blockDim.x + threadIdx.x;
    if (i < N) deg[i] = 1.0f;                 // self loop contributes 1 to degree
    if (i < B * HID) zg[i] = 0.0f;
    if (i < B) { segmax[i] = -__builtin_inff(); denom[i] = 0.0f; }
}

// ---------------------------------------------------------------- fused GEMM:
//   h_lin = x_sem @ W_gcn           [N,64]
//   z_sem = x_sem @ W_ps + b_ps     [N,32]  (written straight to output)
// One wave computes a 16-row tile; 6 column tiles (4 for W_gcn, 2 for W_ps).
// W is staged in LDS pre-paired (k even: {W[k][c],W[k+1][c]}) for b64 loads.
__global__ void k_gemm1(const float* __restrict__ x,
                        const float* __restrict__ Wg, const float* __restrict__ Wp,
                        const float* __restrict__ bps,
                        float* __restrict__ h_lin, float* __restrict__ z_sem, int N) {
    extern __shared__ float sW[];             // (SEM/2)*96 pairs = 147456 bytes
    const int tid = threadIdx.x;
    for (int p = tid; p < (SEM / 2) * 96; p += blockDim.x) {
        int pairIdx = p / 96, c = p % 96, kk = pairIdx * 2;
        float w0, w1;
        if (c < HID) { w0 = Wg[kk * HID + c];            w1 = Wg[(kk + 1) * HID + c]; }
        else { int cc = c - HID; w0 = Wp[kk * ALIGN + cc]; w1 = Wp[(kk + 1) * ALIGN + cc]; }
        sW[2 * p] = w0; sW[2 * p + 1] = w1;
    }
    __syncthreads();

    const int wave = tid >> 5, lane = tid & 31;
    const int half = lane >> 4, m = lane & 15;
    const int rowBase = blockIdx.x * 128 + wave * 16;
    const int row = rowBase + m;
    const int rowC = row < N ? row : N - 1;
    const float* xr = x + (size_t)rowC * SEM + 2 * half;

    v8f acc[6] = {};
    const v2f* sWp = (const v2f*)sW;
    for (int k = 0; k < SEM; k += 4) {
        v2f a = *(const v2f*)(xr + k);
        const v2f* bp = sWp + ((k >> 1) + half) * 96;
#pragma unroll
        for (int t = 0; t < 6; ++t) {
            v2f b = bp[t * 16 + m];
            acc[t] = __builtin_amdgcn_wmma_f32_16x16x4_f32(
                false, a, false, b, (short)0, acc[t], false, false);
        }
    }
#pragma unroll
    for (int t = 0; t < 6; ++t) {
#pragma unroll
        for (int r = 0; r < 8; ++r) {
            int orow = rowBase + 8 * half + r;
            if (orow < N) {
                if (t < 4) {
                    h_lin[(size_t)orow * HID + t * 16 + m] = acc[t][r];
                } else {
                    int c = (t - 4) * 16 + m;
                    z_sem[(size_t)orow * ALIGN + c] = acc[t][r] + bps[c];
                }
            }
        }
    }
}

// ---------------------------------------------------------------- degree
__global__ void k_deg(const int* __restrict__ ei, float* __restrict__ deg, int E) {
    int e = blockIdx.x * blockDim.x + threadIdx.x;
    if (e < E) atomicAdd(&deg[ei[E + e]], 1.0f);   // target = col = edge_index[1]
}

// dinv[i] = rsqrt(deg[i]); agg[i][:] = h_lin[i][:] * dinv^2  (self-loop term)
// block = 4 nodes x 64 features; barrier between deg read and dinv overwrite
__global__ void k_self(const float* __restrict__ h_lin, float* __restrict__ dinv,
                       float* __restrict__ agg, int N) {
    int node = blockIdx.x * 4 + (threadIdx.x >> 6);
    int f = threadIdx.x & 63;
    bool ok = node < N;
    float dg = ok ? dinv[node] : 1.0f;
    float hv = ok ? h_lin[(size_t)node * HID + f] : 0.0f;
    float dv = rsqrtf(dg);
    __syncthreads();
    if (ok) {
        if (f == 0) dinv[node] = dv;
        agg[(size_t)node * HID + f] = hv * dv * dv;
    }
}

// ---------------------------------------------------------------- edge scatter
// one wave per edge; lane handles 2 features (float2); L2-resident atomics
__global__ void k_scatter(const int* __restrict__ ei, const float* __restrict__ h_lin,
                          const float* __restrict__ dinv, float* __restrict__ agg, int E) {
    int wave = (int)((blockIdx.x * (size_t)blockDim.x + threadIdx.x) >> 5);
    int lane = threadIdx.x & 31;
    if (wave >= E) return;
    int r = ei[wave], c = ei[E + wave];
    float norm = dinv[r] * dinv[c];
    v2f v = *(const v2f*)(h_lin + (size_t)r * HID + 2 * lane);
    float* dst = agg + (size_t)c * HID + 2 * lane;
    atomicAdd(dst, v.x * norm);
    atomicAdd(dst + 1, v.y * norm);
}

// ---------------------------------------------------------------- post-GCN:
//   h_topo = relu(agg + b_gcn)  (stored in place into hbuf)
//   z_topo = h_topo @ W_pt + b_pt   (WMMA, K=64, 2 col tiles)
__global__ void k_gcn_post(const float* __restrict__ Wpt, const float* __restrict__ bgcn,
                           const float* __restrict__ bpt,
                           float* __restrict__ hbuf, float* __restrict__ z_topo, int N) {
    __shared__ float sW[HID * ALIGN];          // 8 KB, paired layout
    int tid = threadIdx.x;
    for (int p = tid; p < (HID / 2) * ALIGN; p += blockDim.x) {
        int pairIdx = p >> 5, c = p & 31, kk = pairIdx * 2;
        sW[2 * p] = Wpt[kk * ALIGN + c];
        sW[2 * p + 1] = Wpt[(kk + 1) * ALIGN + c];
    }
    __syncthreads();

    int wave = tid >> 5, lane = tid & 31, half = lane >> 4, m = lane & 15;
    int rowBase = blockIdx.x * 128 + wave * 16;
    int row = rowBase + m;
    size_t base = (size_t)(row < N ? row : N - 1) * HID;

    v2f h[16];
#pragma unroll
    for (int s = 0; s < 16; ++s) {
        int c0 = 4 * s + 2 * half;
        v2f v = *(const v2f*)(hbuf + base + c0);
        v.x += bgcn[c0];     v.y += bgcn[c0 + 1];
        v.x = v.x > 0.0f ? v.x : 0.0f;
        v.y = v.y > 0.0f ? v.y : 0.0f;
        h[s] = v;
    }
    if (row < N) {
#pragma unroll
        for (int s = 0; s < 16; ++s)
            *(v2f*)(hbuf + base + 4 * s + 2 * half) = h[s];
    }

    v8f acc[2] = {};
    const v2f* sWp = (const v2f*)sW;
#pragma unroll
    for (int s = 0; s < 16; ++s) {
        const v2f* bp = sWp + (2 * s + half) * ALIGN;
        acc[0] = __builtin_amdgcn_wmma_f32_16x16x4_f32(
            false, h[s], false, bp[m], (short)0, acc[0], false, false);
        acc[1] = __builtin_amdgcn_wmma_f32_16x16x4_f32(
            false, h[s], false, bp[16 + m], (short)0, acc[1], false, false);
    }
#pragma unroll
    for (int t = 0; t < 2; ++t) {
#pragma unroll
        for (int r = 0; r < 8; ++r) {
            int orow = rowBase + 8 * half + r;
            if (orow < N) {
                int c = t * 16 + m;
                z_topo[(size_t)orow * ALIGN + c] = acc[t][r] + bpt[c];
            }
        }
    }
}

// ---------------------------------------------------------------- score MLP
// s_i = leaky_relu([zt|zs||zt-zs|] @ W_s1 + b_s1) @ W_s2 + b_s2  (per node)
__global__ void k_score(const float* __restrict__ zt_g, const float* __restrict__ zs_g,
                        const float* __restrict__ Ws1, const float* __restrict__ bs1,
                        const float* __restrict__ Ws2, const float* __restrict__ bs2,
                        const int* __restrict__ bidx,
                        float* __restrict__ sbuf, float* __restrict__ segmax, int N) {
    int i = blockIdx.x * blockDim.x + threadIdx.x;
    if (i >= N) return;
    float zt[ALIGN], zs[ALIGN];
#pragma unroll
    for (int k = 0; k < ALIGN; ++k) zt[k] = zt_g[(size_t)i * ALIGN + k];
#pragma unroll
    for (int k = 0; k < ALIGN; ++k) zs[k] = zs_g[(size_t)i * ALIGN + k];
    float hj[16];
#pragma unroll
    for (int j = 0; j < 16; ++j) hj[j] = bs1[j];
#pragma unroll
    for (int k = 0; k < ALIGN; ++k) {
        float ck = zt[k];
        const float* wr = Ws1 + k * 16;
#pragma unroll
        for (int j = 0; j < 16; ++j) hj[j] = fmaf(ck, wr[j], hj[j]);
    }
#pragma unroll
    for (int k = 0; k < ALIGN; ++k) {
        float ck = zs[k];
        const float* wr = Ws1 + (ALIGN + k) * 16;
#pragma unroll
        for (int j = 0; j < 16; ++j) hj[j] = fmaf(ck, wr[j], hj[j]);
    }
#pragma unroll
    for (int k = 0; k < ALIGN; ++k) {
        float ck = fabsf(zt[k] - zs[k]);
        const float* wr = Ws1 + (2 * ALIGN + k) * 16;
#pragma unroll
        for (int j = 0; j < 16; ++j) hj[j] = fmaf(ck, wr[j], hj[j]);
    }
    float s = bs2[0];
#pragma unroll
    for (int j = 0; j < 16; ++j) {
        float t = hj[j];
        t = t > 0.0f ? t : 0.01f * t;          // leaky_relu(0.01)
        s = fmaf(t, Ws2[j], s);
    }
    sbuf[i] = s;
    atomicMaxF(&segmax[bidx[i]], s);
}

// ---------------------------------------------------------------- softmax pieces
__global__ void k_exp(float* __restrict__ sbuf, const float* __restrict__ segmax,
                      float* __restrict__ denom, const int* __restrict__ bidx, int N) {
    int i = blockIdx.x * blockDim.x + threadIdx.x;
    if (i >= N) return;
    int b = bidx[i];
    float e = expf(sbuf[i] - segmax[b]);
    sbuf[i] = e;
    atomicAdd(&denom[b], e);
}

// alpha out + weighted pooling z_graph (block = 4 nodes x 64 features)
__global__ void k_alpha(const float* __restrict__ ebuf, const float* __restrict__ denom,
                        const float* __restrict__ htopo, const int* __restrict__ bidx,
                        float* __restrict__ alpha_out, float* __restrict__ zg, int N) {
    int node = blockIdx.x * 4 + (threadIdx.x >> 6);
    int f = threadIdx.x & 63;
    if (node >= N) return;
    int b = bidx[node];
    float a = ebuf[node] / denom[b];
    if (f == 0) alpha_out[node] = a;
    atomicAdd(&zg[b * HID + f], a * htopo[(size_t)node * HID + f]);
}

// ---------------------------------------------------------------- logits
__global__ void k_logits(const float* __restrict__ zg, const float* __restrict__ Wc,
                         const float* __restrict__ bc, float* __restrict__ out, int B) {
    int b = blockIdx.x * blockDim.x + threadIdx.x;
    if (b >= B) return;
    float s = bc[0];
#pragma unroll
    for (int f = 0; f < HID; ++f) s = fmaf(zg[b * HID + f], Wc[f], s);
    out[b] = s;
}

// ---------------------------------------------------------------- launch
extern "C" void kernel_launch(void* const* d_in, const int* in_sizes, int n_in,
                              void* d_out, int out_size, void* d_ws, size_t ws_size,
                              hipStream_t stream) {
    const float* x_sem = (const float*)d_in[0];
    const int* ei      = (const int*)d_in[1];
    const int* bidx    = (const int*)d_in[2];
    const float* Wg  = (const float*)d_in[3];
    const float* bg  = (const float*)d_in[4];
    const float* Wpt = (const float*)d_in[5];
    const float* bpt = (const float*)d_in[6];
    const float* Wps = (const float*)d_in[7];
    const float* bps = (const float*)d_in[8];
    const float* Ws1 = (const float*)d_in[9];
    const float* bs1 = (const float*)d_in[10];
    const float* Ws2 = (const float*)d_in[11];
    const float* bs2 = (const float*)d_in[12];
    const float* Wc  = (const float*)d_in[13];
    const float* bc  = (const float*)d_in[14];

    const int N = in_sizes[2];
    const int E = in_sizes[1] / 2;
    const int B = out_size - 65 * N;          // out = B + N + 32N + 32N

    // output regions: logits[B] | alpha[N] | z_topo[N*32] | z_sem[N*32]
    float* out = (float*)d_out;
    float* o_logits = out;
    float* o_alpha  = out + B;
    float* o_ztopo  = o_alpha + N;
    float* o_zsem   = o_ztopo + (size_t)N * ALIGN;

    // workspace: h_lin[N*64] | hbuf(agg->h_topo)[N*64] | dinv[N] | sbuf[N] | segmax[B] | denom[B] | zg[B*64]
    float* ws     = (float*)d_ws;
    float* h_lin  = ws;
    float* hbuf   = h_lin + (size_t)N * HID;
    float* dinv   = hbuf + (size_t)N * HID;
    float* sbuf   = dinv + N;
    float* segmax = sbuf + N;
    float* denom  = segmax + B;
    float* zg     = denom + B;

    const int T = 256;
    int initN = N > B * HID ? N : B * HID;
    k_init<<<(initN + T - 1) / T, T, 0, stream>>>(dinv, zg, segmax, denom, N, B);

    int gemmBlocks = (N + 127) / 128;         // 8 waves x 16 rows per block
    size_t smem1 = (size_t)(SEM / 2) * 96 * 2 * sizeof(float);  // 147456 B (<320KB/WGP)
    k_gemm1<<<gemmBlocks, T, smem1, stream>>>(x_sem, Wg, Wps, bps, h_lin, o_zsem, N);

    k_deg<<<(E + T - 1) / T, T, 0, stream>>>(ei, dinv, E);
    k_self<<<(N + 3) / 4, T, 0, stream>>>(h_lin, dinv, hbuf, N);
    k_scatter<<<(E + 7) / 8, T, 0, stream>>>(ei, h_lin, dinv, hbuf, E);

    k_gcn_post<<<gemmBlocks, T, 0, stream>>>(Wpt, bg, bpt, hbuf, o_ztopo, N);

    k_score<<<(N + T - 1) / T, T, 0, stream>>>(o_ztopo, o_zsem, Ws1, bs1, Ws2, bs2,
                                               bidx, sbuf, segmax, N);
    k_exp<<<(N + T - 1) / T, T, 0, stream>>>(sbuf, segmax, denom, bidx, N);
    k_alpha<<<(N + 3) / 4, T, 0, stream>>>(sbuf, denom, hbuf, bidx, o_alpha, zg, N);
    k_logits<<<(B + T - 1) / T, T, 0, stream>>>(zg, Wc, bc, o_logits, B);
}